// RelativePositionEncodingWithOuterSum_43078521979515
// MI455X (gfx1250) — compile-verified
//
#include <hip/hip_runtime.h>

typedef __attribute__((ext_vector_type(2))) float v2f;
typedef __attribute__((ext_vector_type(4))) float f4;
typedef __attribute__((ext_vector_type(8))) float v8f;

#define TF_DIM   22
#define C_Z      128
#define RELPOS_K 32
#define NO_BINS  65          // 2*RELPOS_K + 1
#define C4       (C_Z / 4)   // 32 float4 channel groups

// ---------------------------------------------------------------------------
// Stage 1: E = tf @ W + b for both W_i and W_j using V_WMMA_F32_16X16X4_F32.
// One wave32 per 16x16 output tile; K = 22 -> 5 full K=4 steps + 1 tail step.
// Tail lanes with kb >= 22 load a clamped (in-bounds) address and zero the
// value with v_cndmask — no EXEC manipulation anywhere near the WMMAs.
//
// A (16x4 f32, ISA 7.12.2): lanes 0-15 hold M=lane, VGPR0=K0 VGPR1=K1;
//                           lanes 16-31 hold M=lane-16, VGPR0=K2 VGPR1=K3.
// B (4x16 f32): mirrored — lane holds col N=lane&15, half selects K pair.
// C/D (16x16 f32): VGPR r: lanes 0-15 -> (M=r, N=lane); lanes 16-31 -> (M=8+r).
// ---------------------------------------------------------------------------
__global__ void __launch_bounds__(256)
tf_embed_wmma(const float* __restrict__ tf,
              const float* __restrict__ w_i, const float* __restrict__ b_i,
              const float* __restrict__ w_j, const float* __restrict__ b_j,
              float* __restrict__ Ei, float* __restrict__ Ej, int N)
{
    const int lane = threadIdx.x & 31;
    const int wave = blockIdx.x * 8 + (threadIdx.x >> 5);

    const int tilesM      = N >> 4;                 // 64
    const int tilesPerMat = tilesM * (C_Z >> 4);    // 512
    const int which = wave / tilesPerMat;           // 0 -> E_i, 1 -> E_j
    const int t     = wave % tilesPerMat;
    const int m0    = (t >> 3) << 4;                // tile row base
    const int n0    = (t & 7) << 4;                 // tile col base

    const float* __restrict__ W  = which ? w_j : w_i;
    const float* __restrict__ Bv = which ? b_j : b_i;
    float* __restrict__       E  = which ? Ej  : Ei;

    const int l    = lane & 15;
    const int half = lane >> 4;
    const int row  = m0 + l;   // A-fragment row for this lane
    const int col  = n0 + l;   // B-fragment col for this lane

    v8f acc = {};

    // 5 full K=4 steps (covers K = 0..19, all in-bounds: kb+1 <= 19 < 22).
    #pragma unroll
    for (int ks = 0; ks < 5; ++ks) {
        const int kb = ks * 4 + half * 2;
        v2f a, b;
        a.x = tf[row * TF_DIM + kb];
        a.y = tf[row * TF_DIM + kb + 1];
        b.x = W[kb * C_Z + col];
        b.y = W[(kb + 1) * C_Z + col];
        // (neg_a, A, neg_b, B, c_mod, C, reuse_a, reuse_b)
        acc = __builtin_amdgcn_wmma_f32_16x16x4_f32(false, a, false, b,
                                                    (short)0, acc, false, false);
    }

    // Tail step: kb = 20 + half*2 -> half=0 lanes cover K=20,21; half=1 lanes
    // (K=22,23) are zero-padded. Clamped addresses keep every load in-bounds;
    // the zeroing is a per-lane select (v_cndmask), never an EXEC branch.
    {
        const int  kb    = 20 + half * 2;
        const bool live0 = (kb     < TF_DIM);
        const bool live1 = (kb + 1 < TF_DIM);
        const int  kc0   = live0 ? kb     : 0;
        const int  kc1   = live1 ? kb + 1 : 0;

        const float a0 = tf[row * TF_DIM + kc0];
        const float a1 = tf[row * TF_DIM + kc1];
        const float b0 = W[kc0 * C_Z + col];
        const float b1 = W[kc1 * C_Z + col];

        v2f a, b;
        a.x = live0 ? a0 : 0.0f;
        a.y = live1 ? a1 : 0.0f;
        b.x = live0 ? b0 : 0.0f;
        b.y = live1 ? b1 : 0.0f;
        acc = __builtin_amdgcn_wmma_f32_16x16x4_f32(false, a, false, b,
                                                    (short)0, acc, false, false);
    }

    const float bias = Bv[col];
    #pragma unroll
    for (int r = 0; r < 8; ++r) {
        const int orow = m0 + r + 8 * half;
        E[orow * C_Z + col] = acc[r] + bias;
    }
}

// ---------------------------------------------------------------------------
// Stage 2: pair[i][j][c] = (w_rel[clip(i-j)] + b_rel + E_i[i])[c] + E_j[j][c]
// One block per i. The 65 relpos rows (with b_rel and E_i[i] pre-folded) live
// in LDS (33 KB of the 320 KB WGP pool). Inner loop streams 128-bit NT stores
// — this is the 512 MiB HBM-write-bound part (~23 us floor at 23.3 TB/s).
// ---------------------------------------------------------------------------
__global__ void __launch_bounds__(256)
pair_outer_sum(const f4* __restrict__ Ei4, const f4* __restrict__ Ej4,
               const f4* __restrict__ wrel4, const f4* __restrict__ brel4,
               f4* __restrict__ out4, int N)
{
    __shared__ f4 ldsR[NO_BINS * C4];   // 65 * 32 * 16B = 33280 B

    const int i  = blockIdx.x;
    const int c4 = threadIdx.x & 31;    // channel float4 group
    const int jl = threadIdx.x >> 5;    // 0..7 (j sub-lane)

    // Fold b_rel + E_i[i] into every relpos row once per block.
    const f4 base = brel4[c4] + Ei4[i * C4 + c4];
    for (int k = jl; k < NO_BINS; k += 8)
        ldsR[k * C4 + c4] = wrel4[k * C4 + c4] + base;
    __syncthreads();

    const size_t rowBase = (size_t)i * (size_t)N * C4;
    for (int j = jl; j < N; j += 8) {
        // Prefetch E_j a few iterations ahead (L2-resident stream).
        if (j + 64 < N)
            __builtin_prefetch(&Ej4[(j + 64) * C4 + c4], 0, 1);

        int d = i - j;
        d = (d < -RELPOS_K) ? -RELPOS_K : (d > RELPOS_K ? RELPOS_K : d);
        d += RELPOS_K;

        const f4 v = ldsR[d * C4 + c4] + Ej4[j * C4 + c4];
        __builtin_nontemporal_store(v, &out4[rowBase + (size_t)j * C4 + c4]);
    }
}

// ---------------------------------------------------------------------------
extern "C" void kernel_launch(void* const* d_in, const int* in_sizes, int n_in,
                              void* d_out, int out_size, void* d_ws, size_t ws_size,
                              hipStream_t stream)
{
    const float* tf   = (const float*)d_in[0];
    const float* w_i  = (const float*)d_in[1];
    const float* b_i  = (const float*)d_in[2];
    const float* w_j  = (const float*)d_in[3];
    const float* b_j  = (const float*)d_in[4];
    const float* wrel = (const float*)d_in[5];
    const float* brel = (const float*)d_in[6];

    const int N = in_sizes[0] / TF_DIM;     // B = 1, N = 1024

    float* Ei = (float*)d_ws;               // N*C_Z floats
    float* Ej = Ei + (size_t)N * C_Z;       // N*C_Z floats (1 MiB total ws)

    // Stage 1: 2 * (N/16) * 8 tiles, 8 waves/block -> 2*(N/16) blocks.
    const int tilesM = N / 16;
    tf_embed_wmma<<<dim3(tilesM * 2), 256, 0, stream>>>(
        tf, w_i, b_i, w_j, b_j, Ei, Ej, N);

    // Stage 2: one block per row i.
    pair_outer_sum<<<dim3(N), 256, 0, stream>>>(
        (const f4*)Ei, (const f4*)Ej, (const f4*)wrel, (const f4*)brel,
        (f4*)d_out, N);
}